// SAGE_74594991997216
// MI455X (gfx1250) — compile-verified
//
#include <hip/hip_runtime.h>
#include <hip/hip_bf16.h>

typedef __attribute__((ext_vector_type(2))) float v2f;
typedef __attribute__((ext_vector_type(8))) float v8f;

__device__ __forceinline__ void atomAddF(float* p, float v) {
    __hip_atomic_fetch_add(p, v, __ATOMIC_RELAXED, __HIP_MEMORY_SCOPE_AGENT);
}

// ---------------------------------------------------------------------------
// GEMM: out[M x N] (+)= (A[M x 128] * rowscale) @ W[N x 128]^T (+ bias)
// N = NT*16 (NT = 8 or 4). K fixed at 128.
// One 256-thread block = 8 waves; wave w computes rows [blk*128 + w*16, +16)
// across all N columns using V_WMMA_F32_16X16X4_F32 (full fp32 precision).
// W staged in LDS with XOR swizzle (k ^ ((n&15)<<2)) -> conflict-free B reads,
// exactly 64KB for NT=8.
// ---------------------------------------------------------------------------
template <int NT>
__global__ __launch_bounds__(256)
void gemm_wmma_f32(const float* __restrict__ A, const float* __restrict__ rowscale,
                   const float* __restrict__ W, const float* __restrict__ bias,
                   float* __restrict__ out, int M, int accumulate)
{
    constexpr int N = NT * 16;
    __shared__ float sW[N * 128];

    const int tid = threadIdx.x;
    // Stage W (N x 128, row-major) into LDS, swizzled.
    for (int i = tid; i < N * 32; i += 256) {
        int n  = i >> 5;
        int k4 = (i & 31) << 2;
        int ks = k4 ^ ((n & 15) << 2);
        *(float4*)&sW[n * 128 + ks] = *(const float4*)(W + n * 128 + k4);
    }
    __syncthreads();

    const int wave = tid >> 5;
    const int lane = tid & 31;
    const int m16  = lane & 15;          // row-in-tile (A) / col-in-tile (B,C)
    const int kh   = (lane >> 4) << 1;   // 0 or 2: K sub-offset for A/B frags
    const int row0 = blockIdx.x * 128 + wave * 16;
    if (row0 >= M) return;               // M % 16 == 0, wave-granular tail

    const float ascale = rowscale ? rowscale[row0 + m16] : 1.0f;

    // C/D layout: VGPR r -> row (row0 + r + 8*half), col (n0 + m16)
    v8f acc[NT];
#pragma unroll
    for (int nt = 0; nt < NT; ++nt) {
        v8f c = {0.f, 0.f, 0.f, 0.f, 0.f, 0.f, 0.f, 0.f};
        if (accumulate) {
            const float* o = out + (row0 + kh * 4) * N + nt * 16 + m16;
#pragma unroll
            for (int r = 0; r < 8; ++r) c[r] = o[r * N];
        }
        if (bias) {
            float bv = bias[nt * 16 + m16];
#pragma unroll
            for (int r = 0; r < 8; ++r) c[r] += bv;
        }
        acc[nt] = c;
    }

    // A frag (16x4 fp32): lane<16 -> (M=m16, K=kk..kk+1); lane>=16 -> K=kk+2..kk+3
    const float* aBase = A + (row0 + m16) * 128 + kh;
#pragma unroll 4
    for (int kk = 0; kk < 128; kk += 4) {
        v2f a = *(const v2f*)(aBase + kk);
        a *= ascale;
#pragma unroll
        for (int nt = 0; nt < NT; ++nt) {
            // B frag (4x16 fp32): B(k,n) = W[n][k]; same striping as A.
            const float* bp = &sW[(nt * 16 + m16) * 128 + ((kk + kh) ^ (m16 << 2))];
            v2f b = *(const v2f*)bp;
            acc[nt] = __builtin_amdgcn_wmma_f32_16x16x4_f32(
                false, a, false, b, (short)0, acc[nt], false, false);
        }
    }

#pragma unroll
    for (int nt = 0; nt < NT; ++nt) {
        float* o = out + (row0 + kh * 4) * N + nt * 16 + m16;
#pragma unroll
        for (int r = 0; r < 8; ++r) o[r * N] = acc[nt][r];
    }
}

// ---------------------------------------------------------------------------
// Edge scatter-add: one wave per edge; 32 lanes x float4 = 128 channels.
// agg[dst] += H[src]   (float atomics -> global_atomic_add_f32)
// ---------------------------------------------------------------------------
__global__ __launch_bounds__(256)
void scatter_edges(const float* __restrict__ H, const int* __restrict__ src,
                   const int* __restrict__ dst, float* __restrict__ agg, int E)
{
    int e = blockIdx.x * 8 + (threadIdx.x >> 5);
    if (e >= E) return;
    int lane = threadIdx.x & 31;
    int s = src[e];
    int d = dst[e];
    float4 v = *(const float4*)(H + (size_t)s * 128 + lane * 4);
    float* o = agg + (size_t)d * 128 + lane * 4;
    atomAddF(o + 0, v.x);
    atomAddF(o + 1, v.y);
    atomAddF(o + 2, v.z);
    atomAddF(o + 3, v.w);
}

__global__ __launch_bounds__(256)
void degree_kernel(const int* __restrict__ dst, float* __restrict__ deg, int E)
{
    int e = blockIdx.x * blockDim.x + threadIdx.x;
    if (e < E) atomAddF(&deg[dst[e]], 1.0f);
}

__global__ __launch_bounds__(256)
void invdeg_kernel(float* __restrict__ deg, int n)
{
    int i = blockIdx.x * blockDim.x + threadIdx.x;
    if (i < n) deg[i] = 1.0f / fmaxf(deg[i], 1.0f);
}

// ---------------------------------------------------------------------------
// Fused LayerNorm + ReLU over rows of 128. One wave32 per row, float4/lane,
// cross-lane reductions via __shfl_xor (wave32-correct).
// ---------------------------------------------------------------------------
__global__ __launch_bounds__(256)
void ln_relu_kernel(const float* __restrict__ H, const float* __restrict__ g,
                    const float* __restrict__ b, float* __restrict__ out, int M)
{
    int row = blockIdx.x * 8 + (threadIdx.x >> 5);
    if (row >= M) return;
    int lane = threadIdx.x & 31;
    float4 v = *(const float4*)(H + (size_t)row * 128 + lane * 4);
    float s = v.x + v.y + v.z + v.w;
#pragma unroll
    for (int off = 16; off > 0; off >>= 1) s += __shfl_xor(s, off, 32);
    float mu = s * (1.0f / 128.0f);
    float dx = v.x - mu, dy = v.y - mu, dz = v.z - mu, dw = v.w - mu;
    float q = dx * dx + dy * dy + dz * dz + dw * dw;
#pragma unroll
    for (int off = 16; off > 0; off >>= 1) q += __shfl_xor(q, off, 32);
    float rstd = rsqrtf(q * (1.0f / 128.0f) + 1e-5f);
    float4 gg = *(const float4*)(g + lane * 4);
    float4 bb = *(const float4*)(b + lane * 4);
    float4 o;
    o.x = fmaxf(dx * rstd * gg.x + bb.x, 0.0f);
    o.y = fmaxf(dy * rstd * gg.y + bb.y, 0.0f);
    o.z = fmaxf(dz * rstd * gg.z + bb.z, 0.0f);
    o.w = fmaxf(dw * rstd * gg.w + bb.w, 0.0f);
    *(float4*)(out + (size_t)row * 128 + lane * 4) = o;
}

// ---------------------------------------------------------------------------
extern "C" void kernel_launch(void* const* d_in, const int* in_sizes, int n_in,
                              void* d_out, int out_size, void* d_ws, size_t ws_size,
                              hipStream_t stream)
{
    const float* x     = (const float*)d_in[0];
    const int*   ei    = (const int*)d_in[1];   // [2, E]
    const float* W_ne  = (const float*)d_in[2];
    const float* b_ne  = (const float*)d_in[3];
    const float* W_l   = (const float*)d_in[4];
    const float* b_l   = (const float*)d_in[5];
    const float* W_r   = (const float*)d_in[6];
    const float* ln_g  = (const float*)d_in[7];
    const float* ln_b  = (const float*)d_in[8];
    const float* W_out = (const float*)d_in[9];
    const float* b_out = (const float*)d_in[10];

    const int M = in_sizes[0] / 128;   // 100000 nodes
    const int E = in_sizes[1] / 2;     // 1600000 edges
    const int* src = ei;
    const int* dst = ei + E;

    const size_t MH = (size_t)M * 128;
    float* invdeg = (float*)d_ws;
    float* bufA = invdeg + (((size_t)M + 511) & ~(size_t)511);
    float* bufB = bufA + MH;
    float* bufC = bufB + MH;

    dim3 blk(256);
    const int gemmGrid     = (M + 127) / 128;
    const int edgeWaveGrid = (E + 7) / 8;
    const int rowWaveGrid  = (M + 7) / 8;

    // --- degrees -> inv_deg (in place) ---
    hipMemsetAsync(invdeg, 0, (size_t)M * sizeof(float), stream);
    degree_kernel<<<(E + 255) / 256, blk, 0, stream>>>(dst, invdeg, E);
    invdeg_kernel<<<(M + 255) / 256, blk, 0, stream>>>(invdeg, M);

    // --- h0 = x @ W_ne^T + b_ne -> bufA ---
    gemm_wmma_f32<8><<<gemmGrid, blk, 0, stream>>>(x, nullptr, W_ne, b_ne, bufA, M, 0);

    // --- agg1 = segment_sum(h0[src] -> dst) -> bufB ---
    hipMemsetAsync(bufB, 0, MH * sizeof(float), stream);
    scatter_edges<<<edgeWaveGrid, blk, 0, stream>>>(bufA, src, dst, bufB, E);

    // --- h1 = (agg1 * inv_deg) @ W_l^T + b_l + h0 @ W_r^T -> bufC ---
    gemm_wmma_f32<8><<<gemmGrid, blk, 0, stream>>>(bufB, invdeg, W_l, b_l, bufC, M, 0);
    gemm_wmma_f32<8><<<gemmGrid, blk, 0, stream>>>(bufA, nullptr, W_r, nullptr, bufC, M, 1);

    // --- z = relu(LN(h1)) -> bufA ---
    ln_relu_kernel<<<rowWaveGrid, blk, 0, stream>>>(bufC, ln_g, ln_b, bufA, M);

    // --- agg2 = segment_sum(z[src] -> dst) -> bufB ---
    hipMemsetAsync(bufB, 0, MH * sizeof(float), stream);
    scatter_edges<<<edgeWaveGrid, blk, 0, stream>>>(bufA, src, dst, bufB, E);

    // --- h2 = h1 + (agg2 * inv_deg) @ W_l^T + b_l + z @ W_r^T  (acc into bufC) ---
    gemm_wmma_f32<8><<<gemmGrid, blk, 0, stream>>>(bufB, invdeg, W_l, b_l, bufC, M, 1);
    gemm_wmma_f32<8><<<gemmGrid, blk, 0, stream>>>(bufA, nullptr, W_r, nullptr, bufC, M, 1);

    // --- h3 = relu(LN(h2)) -> bufA ---
    ln_relu_kernel<<<rowWaveGrid, blk, 0, stream>>>(bufC, ln_g, ln_b, bufA, M);

    // --- out = h3 @ W_out^T + b_out  (N = 64) ---
    gemm_wmma_f32<4><<<gemmGrid, blk, 0, stream>>>(bufA, nullptr, W_out, b_out,
                                                   (float*)d_out, M, 0);
}